// ResidualEncoder_86595130622619
// MI455X (gfx1250) — compile-verified
//
#include <hip/hip_runtime.h>
#include <math.h>

// ---------------------------------------------------------------------------
// Problem constants (from reference)
// ---------------------------------------------------------------------------
#define IDIM 80
#define HDIM 512
#define H3   1536      // 3*HDIM
#define ODIM 128
#define G    5
#define PDIM 1285      // G*2*ODIM + G
#define BB   512       // batch (= scan length)
#define TT   512

#define NWG      12          // persistent workgroups for the GRU chain
#define ROWS_WG  128         // H3 / NWG rows of W_hh per WG
#define WHH_STRIDE 516       // padded LDS row stride (2064 B = 129*16 -> 16B aligned rows)

// ---------------------------------------------------------------------------
// CDNA5 WMMA: D = A(16x4) * B(4x16) + C, fp32 everywhere.
// A layout (32-bit, 16x4): lanes 0-15 -> M=lane, VGPR{0,1} = K{0,1};
//                          lanes 16-31 -> M=lane-16, VGPR{0,1} = K{2,3}.
// B mirrored: lane gives N, half-wave gives K pair.
// C/D: 8 VGPRs, lanes 0-15 rows 0-7, lanes 16-31 rows 8-15.
// ---------------------------------------------------------------------------
typedef float v2f __attribute__((ext_vector_type(2)));
typedef float v8f __attribute__((ext_vector_type(8)));

__device__ __forceinline__ v8f wmma4(v2f a, v2f b, v8f c) {
  return __builtin_amdgcn_wmma_f32_16x16x4_f32(
      /*neg_a=*/false, a, /*neg_b=*/false, b,
      /*c_mod=*/(short)0, c, /*reuse_a=*/false, /*reuse_b=*/false);
}

// ---------------------------------------------------------------------------
// Async Global -> LDS (CDNA5 GLOBAL_LOAD_ASYNC_TO_LDS_B128, ASYNCcnt).
// Each active lane moves 16 bytes straight into LDS without touching VGPRs.
// Builtin signature (per hipcc diagnostic): pointers to 16-byte int vectors,
// global source in addrspace(1), LDS destination in addrspace(3).
// ---------------------------------------------------------------------------
#if __has_builtin(__builtin_amdgcn_global_load_async_to_lds_b128) && \
    __has_builtin(__builtin_amdgcn_s_wait_asynccnt)
#define HAVE_ASYNC_LDS 1
typedef int i32x4 __attribute__((__vector_size__(16)));
typedef __attribute__((address_space(1))) i32x4 g_i32x4;
typedef __attribute__((address_space(3))) i32x4 l_i32x4;
__device__ __forceinline__ void async_copy16(const float* g, float* l) {
  __builtin_amdgcn_global_load_async_to_lds_b128((g_i32x4*)g, (l_i32x4*)l,
                                                 /*offset=*/0, /*cpol=*/0);
}
__device__ __forceinline__ void async_wait0() {
  __builtin_amdgcn_s_wait_asynccnt(0);
}
#else
#define HAVE_ASYNC_LDS 0
__device__ __forceinline__ void async_copy16(const float* g, float* l) {
  l[0] = g[0]; l[1] = g[1]; l[2] = g[2]; l[3] = g[3];
}
__device__ __forceinline__ void async_wait0() {}
#endif

// ---------------------------------------------------------------------------
// Kernel 0: init barrier counter
// ---------------------------------------------------------------------------
__global__ void k_init(unsigned* bar) { *bar = 0u; }

// ---------------------------------------------------------------------------
// Kernel 1: xlast[b][g] = sum_f ys[b,511,f] * W_ih[g,f] + b_ih[g]
// M=512 (b) x N=1536 (g), K=80. One wave per 16x16 tile, fp32 WMMA.
// grid: 768 blocks x 128 thr = 3072 waves = 32*96 tiles exactly.
// ---------------------------------------------------------------------------
__global__ void k_xlast(const float* __restrict__ ys,
                        const float* __restrict__ W_ih,
                        const float* __restrict__ b_ih,
                        float* __restrict__ xlast) {
  const int wave = (blockIdx.x * blockDim.x + threadIdx.x) >> 5;
  const int lane = threadIdx.x & 31;
  const int tm = wave & 31;          // 32 M-tiles
  const int tn = wave >> 5;          // 96 N-tiles
  const int l15 = lane & 15;
  const int hi  = lane >> 4;
  const int m = tm * 16 + l15;       // batch row
  const int n = tn * 16 + l15;       // gate column
  const float* arow = ys + ((size_t)m * TT + (TT - 1)) * IDIM + 2 * hi;
  const float* brow = W_ih + (size_t)n * IDIM + 2 * hi;
  v8f c = {0.f, 0.f, 0.f, 0.f, 0.f, 0.f, 0.f, 0.f};
#pragma unroll
  for (int k0 = 0; k0 < IDIM; k0 += 4) {
    v2f a  = *(const v2f*)(arow + k0);
    v2f bv = *(const v2f*)(brow + k0);
    c = wmma4(a, bv, c);
  }
  const float bias = b_ih[n];
  const int row0 = tm * 16 + hi * 8;
#pragma unroll
  for (int r = 0; r < 8; ++r)
    xlast[(size_t)(row0 + r) * H3 + n] = c[r] + bias;
}

// ---------------------------------------------------------------------------
// Kernel 2: persistent GRU chain (single live chain t=511, 512 steps).
// 12 WGs x 512 threads. Each WG: 128-row slice of W_hh in LDS (async-loaded),
// private full copy of h in LDS, double-buffered LDS staging for xlast rows
// (async-prefetched one step ahead, hidden behind the inter-WGP barrier).
// Per step:
//   phase A: gh slice = Whh_slice @ h  (4 threads/row, shfl reduce)
//            -> double-buffered global gh; wait own ASYNCcnt; spin barrier
//   phase B: every WG redundantly computes full h_new from gh + LDS x-row,
//            WG0 writes hist[b]; issue async prefetch of x-row b+1.
// ---------------------------------------------------------------------------
__global__ void k_gru(const float* __restrict__ W_hh,
                      const float* __restrict__ b_hh,
                      const float* __restrict__ xlast,
                      float* __restrict__ hist,
                      float* __restrict__ gh_g,      // 2 * 1536 floats
                      unsigned* __restrict__ bar) {
  extern __shared__ float smem[];
  float* whh  = smem;                          // ROWS_WG * WHH_STRIDE
  float* hbuf = smem + ROWS_WG * WHH_STRIDE;   // HDIM
  float* xbuf = hbuf + HDIM;                   // 2 * H3 (double buffer)
  const int wg  = blockIdx.x;                  // 0..11
  const int tid = threadIdx.x;                 // 0..511

  // one-time async load of this WG's W_hh slice into LDS (16B per call/lane)
  for (int idx = tid; idx < ROWS_WG * (HDIM / 4); idx += 512) {
    const int r  = idx >> 7;          // / (HDIM/4)
    const int k4 = idx & 127;         // % (HDIM/4)
    async_copy16(W_hh + (size_t)(wg * ROWS_WG + r) * HDIM + 4 * k4,
                 whh + r * WHH_STRIDE + 4 * k4);
  }
  // prefetch xlast row 0 into xbuf[0]
  if (tid < H3 / 4)
    async_copy16(xlast + 4 * tid, xbuf + 4 * tid);
  hbuf[tid] = 0.f;   // h0 = 0
  async_wait0();
  __syncthreads();

  const int r = tid >> 2, q = tid & 3;         // 4 threads per output row
  const int grow = wg * ROWS_WG + r;
  const float bias = b_hh[grow];

  for (int b = 0; b < BB; ++b) {
    float* ghw = gh_g + (b & 1) * H3;
    const float* xb = xbuf + (b & 1) * H3;
    // ---- phase A: mat-vec slice, conflict-free k = 4*i + q ----
    float acc = 0.f;
#pragma unroll 8
    for (int i = 0; i < HDIM / 4; ++i)
      acc = fmaf(whh[r * WHH_STRIDE + 4 * i + q], hbuf[4 * i + q], acc);
    acc += __shfl_xor(acc, 1);
    acc += __shfl_xor(acc, 2);
    if (q == 0) ghw[grow] = acc + bias;

    // drain this wave's async queue so LDS x-row is valid after the WG sync
    async_wait0();

    // ---- global barrier across the 12 persistent WGs ----
    __syncthreads();
    if (tid == 0) {
      __threadfence();                       // release gh stores
      atomicAdd(bar, 1u);
      const unsigned target = (unsigned)NWG * (unsigned)(b + 1);
      while (atomicAdd(bar, 0u) < target) { __builtin_amdgcn_s_sleep(1); }
    }
    __syncthreads();
    __threadfence();                         // acquire other WGs' gh

    // ---- async prefetch of next x-row into the other LDS buffer ----
    // (buffer (b+1)&1 was last read at step b-1, sealed by that step's sync)
    if (b + 1 < BB && tid < H3 / 4)
      async_copy16(xlast + (size_t)(b + 1) * H3 + 4 * tid,
                   xbuf + ((b + 1) & 1) * H3 + 4 * tid);

    // ---- phase B: full gate update (redundant per WG), j = tid ----
    const int j = tid;
    const float xr = xb[j];
    const float xz = xb[HDIM + j];
    const float xn = xb[2 * HDIM + j];
    const float hr = ghw[j];
    const float hz = ghw[HDIM + j];
    const float hn = ghw[2 * HDIM + j];
    const float rg = 1.f / (1.f + expf(-(xr + hr)));
    const float zg = 1.f / (1.f + expf(-(xz + hz)));
    const float ng = tanhf(xn + rg * hn);
    const float hnew = (1.f - zg) * ng + zg * hbuf[j];
    if (wg == 0) hist[(size_t)b * HDIM + j] = hnew;
    hbuf[j] = hnew;
    __syncthreads();                         // before next phase A reads hbuf
  }
}

// ---------------------------------------------------------------------------
// Kernel 3: params = hist(512x512) @ W_proj^T (512->1285) + b_proj
// M=512, N padded to 81*16=1296 with store guards, K=512. fp32 WMMA.
// grid: 648 blocks x 128 thr = 2592 waves = 32*81 tiles exactly.
// ---------------------------------------------------------------------------
__global__ void k_params(const float* __restrict__ hist,
                         const float* __restrict__ W_proj,
                         const float* __restrict__ b_proj,
                         float* __restrict__ params) {
  const int wave = (blockIdx.x * blockDim.x + threadIdx.x) >> 5;
  const int lane = threadIdx.x & 31;
  const int tm = wave & 31;            // 32 M-tiles
  const int tn = wave >> 5;            // 81 N-tiles
  const int l15 = lane & 15;
  const int hi  = lane >> 4;
  const int m = tm * 16 + l15;
  const int n = tn * 16 + l15;
  const int nc = (n < PDIM) ? n : (PDIM - 1);   // clamp load row for padding
  const float* arow = hist  + (size_t)m  * HDIM + 2 * hi;
  const float* brow = W_proj + (size_t)nc * HDIM + 2 * hi;
  v8f c = {0.f, 0.f, 0.f, 0.f, 0.f, 0.f, 0.f, 0.f};
#pragma unroll 8
  for (int k0 = 0; k0 < HDIM; k0 += 4) {
    __builtin_prefetch(brow + k0 + 128, 0, 0);   // global_prefetch_b8
    v2f a  = *(const v2f*)(arow + k0);
    v2f bv = *(const v2f*)(brow + k0);
    c = wmma4(a, bv, c);
  }
  if (n < PDIM) {
    const float bias = b_proj[n];
    const int row0 = tm * 16 + hi * 8;
#pragma unroll
    for (int r = 0; r < 8; ++r)
      params[(size_t)(row0 + r) * PDIM + n] = c[r] + bias;
  }
}

// ---------------------------------------------------------------------------
// Kernel 4: mixture sampling (matches reference's quirky var indexing)
// out[b,o] = sum_i params[b,1280+i] * (mu_i + var_i * eps[i,b,o])
//   mu_i  = params[b, i*128 + o]
//   var_i = params[b, 2*i*128 + o] + 1e-6
// ---------------------------------------------------------------------------
__global__ void k_sample(const float* __restrict__ params,
                         const float* __restrict__ eps,
                         float* __restrict__ out) {
  const int idx = blockIdx.x * blockDim.x + threadIdx.x;   // 512*128
  const int b = idx >> 7, o = idx & 127;
  const float* pr = params + (size_t)b * PDIM;
  float acc = 0.f;
#pragma unroll
  for (int i = 0; i < G; ++i) {
    const float mu  = pr[i * ODIM + o];
    const float var = pr[2 * i * ODIM + o] + 1e-6f;
    const float e   = eps[((size_t)i * BB + b) * ODIM + o];
    const float w   = pr[2 * G * ODIM + i];
    acc += w * (mu + var * e);
  }
  out[(size_t)b * ODIM + o] = acc;
}

// ---------------------------------------------------------------------------
// Host launcher
// ---------------------------------------------------------------------------
extern "C" void kernel_launch(void* const* d_in, const int* in_sizes, int n_in,
                              void* d_out, int out_size, void* d_ws, size_t ws_size,
                              hipStream_t stream) {
  const float* ys     = (const float*)d_in[0];
  const float* W_ih   = (const float*)d_in[1];
  const float* W_hh   = (const float*)d_in[2];
  const float* b_ih   = (const float*)d_in[3];
  const float* b_hh   = (const float*)d_in[4];
  const float* W_proj = (const float*)d_in[5];
  const float* b_proj = (const float*)d_in[6];
  const float* eps    = (const float*)d_in[7];
  float* out = (float*)d_out;

  // workspace layout (floats): xlast | hist | gh(2x1536) | params | barrier
  float* xlast  = (float*)d_ws;
  float* hist   = xlast + (size_t)BB * H3;
  float* gh     = hist  + (size_t)BB * HDIM;
  float* params = gh + 2 * H3;
  unsigned* bar = (unsigned*)(params + (size_t)BB * PDIM);

  const int lds_bytes =
      (ROWS_WG * WHH_STRIDE + HDIM + 2 * H3) * (int)sizeof(float); // ~272 KB
  (void)hipFuncSetAttribute((const void*)k_gru,
                            hipFuncAttributeMaxDynamicSharedMemorySize, lds_bytes);

  k_init<<<1, 1, 0, stream>>>(bar);
  k_xlast<<<768, 128, 0, stream>>>(ys, W_ih, b_ih, xlast);          // 3072 waves
  k_gru<<<NWG, 512, lds_bytes, stream>>>(W_hh, b_hh, xlast, hist, gh, bar);
  k_params<<<648, 128, 0, stream>>>(hist, W_proj, b_proj, params);  // 2592 waves
  k_sample<<<(BB * ODIM) / 256, 256, 0, stream>>>(params, eps, out);
}